// AttentionKernel_Position_47502338294174
// MI455X (gfx1250) — compile-verified
//
#include <hip/hip_runtime.h>
#include <math.h>

#define B_ 32
#define D_ 512
#define S_ 4096
#define H_ 512

typedef __attribute__((ext_vector_type(2))) float v2f;
typedef __attribute__((ext_vector_type(8))) float v8f;

// ---------------------------------------------------------------------------
// 1) Sinusoidal positional encoding: pe[s,2i]=sin(s*div_i), pe[s,2i+1]=cos
//    div_i = exp(2i * -ln(10000)/512)
// ---------------------------------------------------------------------------
__global__ void pe_kernel(float* __restrict__ pe) {
    int idx = blockIdx.x * blockDim.x + threadIdx.x;   // 0 .. S_*H_/2-1
    int s = idx >> 8;                                  // 256 (sin,cos) pairs per s
    int i = idx & 255;
    float div = __expf((float)(2 * i) * (-9.210340371976184f / (float)H_));
    float ang = (float)s * div;
    float sv, cv;
    __sincosf(ang, &sv, &cv);
    pe[(size_t)s * H_ + 2 * i]     = sv;
    pe[(size_t)s * H_ + 2 * i + 1] = cv;
}

// ---------------------------------------------------------------------------
// 2) q[b,h] = sum_d Y[b,d,S-1]*W[h,d] + bias[h] + Y[b,h,S-1] + pe[S-1,h]
//    plus c[b] = sum_h bias[h]*q[b,h]  (LDS tree reduction)
//    One 512-thread block per batch b.
// ---------------------------------------------------------------------------
__global__ void q_kernel(const float* __restrict__ Y, const float* __restrict__ W,
                         const float* __restrict__ bias, const float* __restrict__ pe,
                         float* __restrict__ q, float* __restrict__ c) {
    __shared__ float ylast[D_];
    __shared__ float red[512];
    int b = blockIdx.x;
    int h = threadIdx.x;                               // 512 threads
    ylast[h] = Y[(size_t)b * D_ * S_ + (size_t)h * S_ + (S_ - 1)];
    __syncthreads();
    float acc = bias[h];
    const float* wrow = W + (size_t)h * D_;
#pragma unroll 4
    for (int d = 0; d < D_; ++d) acc = fmaf(ylast[d], wrow[d], acc);
    acc += ylast[h] + pe[(size_t)(S_ - 1) * H_ + h];
    q[b * H_ + h] = acc;

    red[h] = bias[h] * acc;
    __syncthreads();
    for (int st = 256; st > 0; st >>= 1) {
        if (h < st) red[h] += red[h + st];
        __syncthreads();
    }
    if (h == 0) c[b] = red[0];
}

// ---------------------------------------------------------------------------
// 3) v[b,d] = sum_h q[b,h]*W[h,d] + q[b,d]   (coalesced: all threads read W row h)
// ---------------------------------------------------------------------------
__global__ void v_kernel(const float* __restrict__ W, const float* __restrict__ q,
                         float* __restrict__ v) {
    __shared__ float qs[H_];
    int b = blockIdx.x;
    int d = threadIdx.x;                               // 512 threads
    qs[d] = q[b * H_ + d];
    __syncthreads();
    float acc = 0.f;
#pragma unroll 4
    for (int h = 0; h < H_; ++h) acc = fmaf(qs[h], W[(size_t)h * D_ + d], acc);
    v[b * D_ + d] = acc + qs[d];
}

// ---------------------------------------------------------------------------
// 4) PEQ[b,s] = sum_h q[b,h]*pe[s,h]  — f32 WMMA 16x16x4 (exact f32 path).
//    One wave per 16(b)x16(s) tile; K-loop over h in steps of 4.
//    A 16x4 f32 layout: lanes 0-15 rows M, VGPR0/1 = K pair (lanes16-31: K+2).
//    B 4x16 f32 layout mirrored (lanes = N cols). C/D: 8 VGPRs, M = r + 8*(lane>=16).
// ---------------------------------------------------------------------------
__global__ void peq_kernel(const float* __restrict__ q, const float* __restrict__ pe,
                           float* __restrict__ peq) {
    int wave = threadIdx.x >> 5;
    int lane = threadIdx.x & 31;
    int s0 = blockIdx.x * 64 + wave * 16;              // 4 waves -> 64 s per block
    int b0 = blockIdx.y * 16;                          // 2 b-tiles cover B=32
    int mn   = lane & 15;
    int half = lane >> 4;

    const float* qrow = q  + (size_t)(b0 + mn) * H_;   // A row m
    const float* prow = pe + (size_t)(s0 + mn) * H_;   // B col n = pe row s0+n

    v8f acc = {};
    for (int h0 = 0; h0 < H_; h0 += 4) {
        int k = h0 + 2 * half;
        v2f a;  a.x  = qrow[k]; a.y  = qrow[k + 1];
        v2f bb; bb.x = prow[k]; bb.y = prow[k + 1];
        acc = __builtin_amdgcn_wmma_f32_16x16x4_f32(
            /*neg_a=*/false, a, /*neg_b=*/false, bb,
            /*c_mod=*/(short)0, acc, /*reuse_a=*/false, /*reuse_b=*/false);
    }
#pragma unroll
    for (int r = 0; r < 8; ++r) {
        int m = r + half * 8;
        peq[(size_t)(b0 + m) * S_ + s0 + mn] = acc[r];
    }
}

// ---------------------------------------------------------------------------
// 5) scores[b,s] = (sum_d Y[b,d,s]*v[b,d] + c[b] + PEQ[b,s]) / sqrt(512)
//    The bandwidth kernel: streams y_history exactly once (256 MB).
//    512 blocks x 256 threads; each block owns a 256-wide s slab for one b.
// ---------------------------------------------------------------------------
__global__ void score_kernel(const float* __restrict__ Y, const float* __restrict__ v,
                             const float* __restrict__ c, const float* __restrict__ peq,
                             float* __restrict__ scores) {
    __shared__ float vs[D_];
    int b = blockIdx.x >> 4;
    int chunk = blockIdx.x & 15;
    int tid = threadIdx.x;                             // 256 threads
    vs[tid]       = v[b * D_ + tid];
    vs[tid + 256] = v[b * D_ + tid + 256];
    __syncthreads();
    int s = chunk * 256 + tid;
    const float* ycol = Y + (size_t)b * D_ * S_ + s;
    float acc = 0.f;
#pragma unroll 4
    for (int d = 0; d < D_; ++d) {
        if (d + 16 < D_) __builtin_prefetch(ycol + (size_t)(d + 16) * S_, 0, 1);
        acc = fmaf(ycol[(size_t)d * S_], vs[d], acc);
    }
    const float inv = 0.04419417382415922f;            // 1/sqrt(512)
    scores[(size_t)b * S_ + s] = (acc + c[b] + peq[(size_t)b * S_ + s]) * inv;
}

// ---------------------------------------------------------------------------
// 6) Row softmax over S=4096 per batch. 32 blocks x 256 threads, 16 vals/thread.
// ---------------------------------------------------------------------------
__global__ void softmax_kernel(const float* __restrict__ scores, float* __restrict__ out) {
    __shared__ float red[256];
    int b = blockIdx.x;
    int tid = threadIdx.x;
    float vals[16];
    float m = -INFINITY;
#pragma unroll
    for (int k = 0; k < 16; ++k) {
        vals[k] = scores[(size_t)b * S_ + k * 256 + tid];
        m = fmaxf(m, vals[k]);
    }
    red[tid] = m;
    __syncthreads();
    for (int st = 128; st > 0; st >>= 1) {
        if (tid < st) red[tid] = fmaxf(red[tid], red[tid + st]);
        __syncthreads();
    }
    m = red[0];
    __syncthreads();
    float sum = 0.f;
#pragma unroll
    for (int k = 0; k < 16; ++k) { vals[k] = __expf(vals[k] - m); sum += vals[k]; }
    red[tid] = sum;
    __syncthreads();
    for (int st = 128; st > 0; st >>= 1) {
        if (tid < st) red[tid] += red[tid + st];
        __syncthreads();
    }
    float inv = 1.0f / red[0];
#pragma unroll
    for (int k = 0; k < 16; ++k)
        out[(size_t)b * S_ + k * 256 + tid] = vals[k] * inv;
}

// ---------------------------------------------------------------------------
extern "C" void kernel_launch(void* const* d_in, const int* in_sizes, int n_in,
                              void* d_out, int out_size, void* d_ws, size_t ws_size,
                              hipStream_t stream) {
    // setup_inputs order: t_current, t_history, y_current, y_history, Wk_w, Wk_b
    const float* Y    = (const float*)d_in[3];
    const float* W    = (const float*)d_in[4];
    const float* bias = (const float*)d_in[5];

    float* ws = (float*)d_ws;
    size_t o = 0;
    float* pe     = ws + o; o += (size_t)S_ * H_;   // 8 MB
    float* q      = ws + o; o += (size_t)B_ * H_;
    float* v      = ws + o; o += (size_t)B_ * D_;
    float* c      = ws + o; o += B_;
    float* peq    = ws + o; o += (size_t)B_ * S_;
    float* scores = ws + o;
    float* out = (float*)d_out;

    pe_kernel     <<<S_ * (H_ / 2) / 256, 256, 0, stream>>>(pe);
    q_kernel      <<<B_, 512, 0, stream>>>(Y, W, bias, pe, q, c);
    v_kernel      <<<B_, 512, 0, stream>>>(W, q, v);
    peq_kernel    <<<dim3(S_ / 64, B_ / 16), 128, 0, stream>>>(q, pe, peq);
    score_kernel  <<<B_ * 16, 256, 0, stream>>>(Y, v, c, peq, scores);
    softmax_kernel<<<B_, 256, 0, stream>>>(scores, out);
}